// BackwardTransformLayer_39393440039485
// MI455X (gfx1250) — compile-verified
//
#include <hip/hip_runtime.h>
#include <stdint.h>

// Problem constants (from reference): N=4096 rows, M=8192 input cols, n=2M output cols.
#define ROWS   4096
#define MCOLS  8192
#define NCOLS  (2 * MCOLS)
#define QT     1024            // input elements per tile
#define NTILE  (MCOLS / QT)    // 8 tiles per row
#define TPB    256             // 8 waves (wave32)

typedef float v4f __attribute__((ext_vector_type(4)));

// ---- CDNA5 async global->LDS copy (Tensor/async data-mover path) ----------
// the LDS destination address comes from a VGPR; the low 32 bits of a generic pointer to a
// __shared__ object are exactly the wave-relative LDS byte offset.
__device__ __forceinline__ void async_ld_b128(const float* lds_dst, const float* gsrc) {
    uint32_t lds_off = (uint32_t)(uintptr_t)lds_dst;
    asm volatile("global_load_async_to_lds_b128 %0, %1, off"
                 :
                 : "v"(lds_off), "v"(gsrc)
                 : "memory");
}
#define WAIT_ASYNC(n) asm volatile("s_wait_asynccnt " #n ::: "memory")

__global__ __launch_bounds__(TPB)
void iwt_synthesis_kernel(const float* __restrict__ details,
                          const float* __restrict__ approx,
                          const float* __restrict__ scaling,
                          const float* __restrict__ scaling_rec,
                          float* __restrict__ out)
{
    // double-buffered tiles: [buf][d/a][QT + 4 halo (+pad to 16B multiple)]
    __shared__ __align__(16) float sm[2][2][QT + 8];

    const int tid = threadIdx.x;
    const int row = blockIdx.x;
    const float* rowD = details + (size_t)row * MCOLS;
    const float* rowA = approx  + (size_t)row * MCOLS;
    float*       rowO = out     + (size_t)row * NCOLS;

    // 8-tap synthesis filters (uniform loads, done once).
    float h[8], g[8];
#pragma unroll
    for (int j = 0; j < 8; ++j) h[j] = scaling[j];
#pragma unroll
    for (int j = 0; j < 8; ++j)
        g[j] = scaling_rec[7 - j] * ((j & 1) ? -1.0f : 1.0f);

    // ---- prologue: stage tile 0 into buffer 0 -----------------------------
    {
        async_ld_b128(&sm[0][0][4 * tid], rowD + 4 * tid);
        async_ld_b128(&sm[0][1][4 * tid], rowA + 4 * tid);
        if ((tid & 31) == 0) {   // one lane per wave: all 8 waves issue 4 asyncs
            async_ld_b128(&sm[0][0][QT], rowD + QT);  // halo (no wrap for tile 0)
            async_ld_b128(&sm[0][1][QT], rowA + QT);
        }
    }

    int buf = 0;
    for (int tile = 0; tile < NTILE; ++tile) {
        const int q0 = tile * QT;

        // ---- stage next tile into the other buffer, then wait for current ----
        if (tile + 1 < NTILE) {
            const int q1 = q0 + QT;
            async_ld_b128(&sm[buf ^ 1][0][4 * tid], rowD + q1 + 4 * tid);
            async_ld_b128(&sm[buf ^ 1][1][4 * tid], rowA + q1 + 4 * tid);
            if ((tid & 31) == 0) {
                const int qh = (q1 + QT) & (MCOLS - 1);   // circular halo
                async_ld_b128(&sm[buf ^ 1][0][QT], rowD + qh);
                async_ld_b128(&sm[buf ^ 1][1][QT], rowA + qh);
            }
            // 4 asyncs/wave just issued for the NEXT tile are allowed to remain
            // outstanding; everything older (current tile) must be complete.
            WAIT_ASYNC(4);
        } else {
            WAIT_ASYNC(0);
        }
        __syncthreads();

        // ---- compute 8 outputs per thread from LDS ---------------------------
        const float* sD = sm[buf][0];
        const float* sA = sm[buf][1];
        const int l = 4 * tid;

        const v4f d0 = ((const v4f*)&sD[l])[0];
        const v4f d1 = ((const v4f*)&sD[l])[1];
        const v4f a0 = ((const v4f*)&sA[l])[0];
        const v4f a1 = ((const v4f*)&sA[l])[1];
        float dv[8] = {d0.x, d0.y, d0.z, d0.w, d1.x, d1.y, d1.z, d1.w};
        float av[8] = {a0.x, a0.y, a0.z, a0.w, a1.x, a1.y, a1.z, a1.w};

        float o[8];
#pragma unroll
        for (int s = 0; s < 4; ++s) {
            float e  = 0.0f;   // out[2q]  : taps g[0,2,4,6] / h[0,2,4,6]
            float od = 0.0f;   // out[2q+1]: taps g[1,3,5,7] / h[1,3,5,7]
#pragma unroll
            for (int t = 0; t < 4; ++t) {
                e  = fmaf(dv[s + t],     g[2 * t],     e);
                e  = fmaf(av[s + t],     h[2 * t],     e);
                od = fmaf(dv[s + 1 + t], g[2 * t + 1], od);
                od = fmaf(av[s + 1 + t], h[2 * t + 1], od);
            }
            o[2 * s]     = e;
            o[2 * s + 1] = od;
        }

        // 8 consecutive outputs -> two 16B non-temporal stores (write-once data).
        v4f w0 = {o[0], o[1], o[2], o[3]};
        v4f w1 = {o[4], o[5], o[6], o[7]};
        v4f* po = (v4f*)(rowO + 2 * q0 + 8 * tid);
        __builtin_nontemporal_store(w0, po);
        __builtin_nontemporal_store(w1, po + 1);

        __syncthreads();   // all reads of sm[buf] done before it is re-staged
        buf ^= 1;
    }
}

extern "C" void kernel_launch(void* const* d_in, const int* in_sizes, int n_in,
                              void* d_out, int out_size, void* d_ws, size_t ws_size,
                              hipStream_t stream) {
    const float* details     = (const float*)d_in[0];
    const float* approximat  = (const float*)d_in[1];
    const float* scaling     = (const float*)d_in[2];
    const float* scaling_rec = (const float*)d_in[3];
    float* out = (float*)d_out;

    dim3 grid(ROWS);
    dim3 block(TPB);
    hipLaunchKernelGGL(iwt_synthesis_kernel, grid, block, 0, stream,
                       details, approximat, scaling, scaling_rec, out);
}